// SimpleRNN_64957085385400
// MI455X (gfx1250) — compile-verified
//
#include <hip/hip_runtime.h>
#include <hip/hip_bf16.h>
#include <math.h>

typedef __attribute__((ext_vector_type(2))) float v2f;
typedef __attribute__((ext_vector_type(8))) float v8f;

#define BB     256      // batch
#define TT     512      // time steps
#define EE     64       // embedding dim
#define HH     256      // hidden dim
#define NOUT   96003    // OUT_LEN * VOCAB = 3 * 32001

#if __has_builtin(__builtin_amdgcn_tanhf)
#define TANHF(v) __builtin_amdgcn_tanhf(v)
#elif __has_builtin(__builtin_amdgcn_tanh_f32)
#define TANHF(v) __builtin_amdgcn_tanh_f32(v)
#else
#define TANHF(v) tanhf(v)
#endif

// ---------------------------------------------------------------------------
// Recurrence kernel (fused embedding + input projection + tanh RNN scan).
// 16 workgroups, each owns 16 batch rows for all 512 steps (no cross-WG sync).
// W_hh lives in LDS for the whole kernel, retiled PER WAVE so the B-fragment
// LDS index has the same form as the A-fragment index (kk-stride 128B) ->
// immediate-offset ds_load_2addr_b64 throughout the fully unrolled K loop.
// Embedding gather is software-pipelined one step ahead; 4 independent WMMA
// accumulators shorten the per-step dependent chain from 80 to ~20.
// ---------------------------------------------------------------------------
__global__ __launch_bounds__(512) void rnn_recurrence_kernel(
    const int*   __restrict__ x,       // [B,T]
    const float* __restrict__ emb,     // [VOCAB,E]
    const float* __restrict__ W_ih,    // [H,E]
    const float* __restrict__ W_hh,    // [H,H]
    const float* __restrict__ b_ih,    // [H]
    const float* __restrict__ b_hh,    // [H]
    float*       __restrict__ hT)      // [B,H] (workspace output)
{
    extern __shared__ char smem_raw[];
    // LDS carve-up (293KB total, <= 320KB WGP LDS):
    v2f*   Wp   = (v2f*)(smem_raw);                      // [16 tile][128 kk][16 c] = 256KB
    v2f*   hp0  = (v2f*)(smem_raw + 262144);             // [128 kk][16 m] = 16KB
    v2f*   hp1  = (v2f*)(smem_raw + 262144 + 16384);     // [128 kk][16 m] = 16KB
    v2f*   ep   = (v2f*)(smem_raw + 262144 + 32768);     // [32 kk][16 m]  = 4KB
    float* bias = (float*)(smem_raw + 262144 + 36864);   // [256] = 1KB

    const int tid  = threadIdx.x;
    const int b0   = blockIdx.x * 16;          // this WG's batch rows
    const int wave = tid >> 5;
    const int lane = tid & 31;
    const int m16  = lane & 15;                // A-side row / B-side local col
    const int half = lane >> 4;                // K sub-chunk selector
    const int n    = wave * 16 + m16;          // global B-side column

    // ---- preload W_hh into LDS, retiled per wave-tile ----
    for (int idx = tid; idx < 128 * 256; idx += 512) {
        const int kk = idx & 127, nn = idx >> 7;
        Wp[(nn >> 4) * 2048 + kk * 16 + (nn & 15)] =
            *(const v2f*)(W_hh + nn * HH + 2 * kk);
    }
    for (int nn = tid; nn < HH; nn += 512) bias[nn] = b_ih[nn] + b_hh[nn];
    {   v2f z; z.x = 0.f; z.y = 0.f;           // h0 = 0
        for (int idx = tid; idx < 128 * 16; idx += 512) hp0[idx] = z;
    }

    // ---- hoist this wave's W_ih B-tile into registers (16 x v2f = 32 VGPRs) ----
    v2f bih[16];
#pragma unroll
    for (int kc = 0; kc < 16; ++kc) {
        bih[kc] = *(const v2f*)(W_ih + n * EE + 4 * kc + 2 * half);
    }
    __syncthreads();

    const v2f* WpT = Wp + wave * 2048;         // this wave's W_hh tile
    v2f* hc = hp0;                             // current h
    v2f* hn = hp1;                             // next h

    // ---- software-pipelined embedding gather: prime step 0 ----
    const int erow  = tid >> 5;                // row 0..15
    const int ecol  = (tid & 31) * 2;          // col pair 0..62
    const int xbase = (b0 + erow) * TT;
    v2f ev = *(const v2f*)(emb + x[xbase] * EE + ecol);

    for (int t = 0; t < TT; ++t) {
        // publish this step's embedding tile (prefetched last iteration)
        ep[(ecol >> 1) * 16 + erow] = ev;
        __syncthreads();

        // issue next step's token load early (latency hides under WMMAs)
        const int tok_next = (t + 1 < TT) ? x[xbase + t + 1] : 0;

        v8f acc[4];
#pragma unroll
        for (int q = 0; q < 4; ++q) acc[q] = (v8f){};

        // input projection: K = 64 (B-fragments from registers)
#pragma unroll
        for (int kc = 0; kc < 16; ++kc) {
            const v2f a = ep[(2 * kc + half) * 16 + m16];
            acc[kc & 1] = __builtin_amdgcn_wmma_f32_16x16x4_f32(
                false, a, false, bih[kc], (short)0, acc[kc & 1], false, false);
        }
        // recurrence: K = 256; A and B indices identical in form -> 2addr LDS
#pragma unroll
        for (int kc = 0; kc < 64; ++kc) {
            const int o = (2 * kc + half) * 16 + m16;
            const v2f a = hc[o];
            const v2f b = WpT[o];
            acc[kc & 3] = __builtin_amdgcn_wmma_f32_16x16x4_f32(
                false, a, false, b, (short)0, acc[kc & 3], false, false);
        }

        // issue next step's embedding-row load (token arrived during WMMAs)
        ev = *(const v2f*)(emb + tok_next * EE + ecol);

        // reduce the 4 partial accumulators
        const v8f s01 = acc[0] + acc[1];
        const v8f s23 = acc[2] + acc[3];
        const v8f asum = s01 + s23;

        // bias + tanh + scatter into next-h (C elem i -> m = i+8*half, same n)
        const float bsum = bias[n];
        float* hnf = (float*)hn;
#pragma unroll
        for (int i = 0; i < 8; ++i) {
            const int m = i + 8 * half;
            hnf[(n >> 1) * 32 + m * 2 + (n & 1)] = TANHF(asum[i] + bsum);
        }
        __syncthreads();
        v2f* tmp = hc; hc = hn; hn = tmp;
    }

    // ---- write final hidden state to workspace ----
    const float* hcf = (const float*)hc;
    for (int idx = tid; idx < 16 * HH; idx += 512) {
        const int m = idx >> 8, nn = idx & 255;
        hT[(b0 + m) * HH + nn] = hcf[(nn >> 1) * 32 + m * 2 + (nn & 1)];
    }
}

// ---------------------------------------------------------------------------
// Head GEMM: logits[256, 96003] = hT @ W_head^T + b_head.
// W_head (98MB) is read once per 16-row batch block (16x) -> non-temporal loads; logits (94MB)
// written once -> non-temporal stores. Each lane streams one W_head row
// contiguously across the K loop. 2 accumulators break the WMMA chain.
// ---------------------------------------------------------------------------
__global__ __launch_bounds__(256) void rnn_head_kernel(
    const float* __restrict__ hT,      // [B,H]
    const float* __restrict__ W_head,  // [NOUT,H]
    const float* __restrict__ b_head,  // [NOUT]
    float*       __restrict__ out)     // [B,NOUT]
{
    __shared__ v2f hp[128 * 16];       // 16KB: hp[kk*16+m] = {h[m][2kk], h[m][2kk+1]}
    const int tid = threadIdx.x;
    const int m0  = blockIdx.y * 16;

    for (int idx = tid; idx < 128 * 16; idx += 256) {
        const int k2 = idx & 127, m = idx >> 7;
        hp[k2 * 16 + m] = *(const v2f*)(hT + (m0 + m) * HH + 2 * k2);
    }
    __syncthreads();

    const int wave = tid >> 5;
    const int lane = tid & 31;
    const int m16  = lane & 15;
    const int half = lane >> 4;
    const long n0  = ((long)blockIdx.x * 8 + wave) * 16;
    if (n0 >= NOUT) return;            // wave-uniform: EXEC stays all-ones

    const int n  = (int)n0 + m16;
    const int nc = n < NOUT ? n : NOUT - 1;     // clamp for safe tail loads
    const float* wrow = W_head + (long)nc * HH;

    v8f acc0 = {}, acc1 = {};
#pragma unroll 8
    for (int kc = 0; kc < 64; kc += 2) {
        const v2f a0 = hp[(2 * kc + half) * 16 + m16];
        const v2f b0 = __builtin_nontemporal_load(
                           (const v2f*)(wrow + 4 * kc + 2 * half));
        acc0 = __builtin_amdgcn_wmma_f32_16x16x4_f32(
                   false, a0, false, b0, (short)0, acc0, false, false);
        const v2f a1 = hp[(2 * (kc + 1) + half) * 16 + m16];
        const v2f b1 = __builtin_nontemporal_load(
                           (const v2f*)(wrow + 4 * (kc + 1) + 2 * half));
        acc1 = __builtin_amdgcn_wmma_f32_16x16x4_f32(
                   false, a1, false, b1, (short)0, acc1, false, false);
    }
    const v8f acc = acc0 + acc1;

    const float bb = b_head[nc];
    if (n < NOUT) {                    // divergence only after all WMMAs
#pragma unroll
        for (int i = 0; i < 8; ++i) {
            const int m = m0 + i + 8 * half;
            __builtin_nontemporal_store(acc[i] + bb, &out[(long)m * NOUT + n]);
        }
    }
}

// ---------------------------------------------------------------------------
extern "C" void kernel_launch(void* const* d_in, const int* in_sizes, int n_in,
                              void* d_out, int out_size, void* d_ws, size_t ws_size,
                              hipStream_t stream) {
    const int*   x      = (const int*)  d_in[0];
    const float* emb    = (const float*)d_in[1];
    const float* W_ih   = (const float*)d_in[2];
    const float* W_hh   = (const float*)d_in[3];
    const float* b_ih   = (const float*)d_in[4];
    const float* b_hh   = (const float*)d_in[5];
    const float* W_head = (const float*)d_in[6];
    const float* b_head = (const float*)d_in[7];
    float* out = (float*)d_out;
    float* hT  = (float*)d_ws;         // 256*256 f32 = 256KB scratch

    const size_t lds_bytes = 262144 + 16384 + 16384 + 4096 + 1024; // 293KB

    rnn_recurrence_kernel<<<dim3(16), dim3(512), lds_bytes, stream>>>(
        x, emb, W_ih, W_hh, b_ih, b_hh, hT);

    const int n_tiles = (NOUT + 15) / 16;          // 6001
    dim3 grid((n_tiles + 7) / 8, BB / 16);         // (751, 16)
    rnn_head_kernel<<<grid, dim3(256), 0, stream>>>(hT, W_head, b_head, out);
}